// TrashNet_54760833024622
// MI455X (gfx1250) — compile-verified
//
#include <hip/hip_runtime.h>
#include <hip/hip_bf16.h>

typedef __attribute__((ext_vector_type(2))) float v2f;
typedef __attribute__((ext_vector_type(8))) float v8f;

#define IN_FEATS 128
#define HID 64

// -----------------------------------------------------------------------------
// Kernel 1: fused dual GEMM  xWself = x @ W_self,  hneigh = x @ W_neigh
// x: [N,128] row-major, W: [128,64] row-major. One block = 16 rows, 4 waves,
// each wave owns one 16-col tile and both weight matrices (2 accumulators).
// K=128 -> 32 x V_WMMA_F32_16X16X4_F32 per accumulator. 100000 % 16 == 0, so
// EXEC is all-1s for every WMMA (no partial tiles).
// -----------------------------------------------------------------------------
__global__ void sage_dual_gemm(const float* __restrict__ x,
                               const float* __restrict__ Wself,
                               const float* __restrict__ Wneigh,
                               float* __restrict__ xWself,
                               float* __restrict__ hneigh) {
    const int lane = threadIdx.x & 31;
    const int wave = threadIdx.x >> 5;      // 0..3 -> N tile
    const int col0 = wave * 16;
    const int row0 = blockIdx.x * 16;
    const int l    = lane & 15;
    const int half = lane >> 4;             // K split across lane halves

    // A fragment source: lane covers row (row0+l), k = 4*kk + 2*half + {0,1}
    const float* arow = x + (size_t)(row0 + l) * IN_FEATS + half * 2;
    // B fragment source: lane covers col (col0+l), k = 4*kk + 2*half + {0,1}
    const float* bs = Wself  + col0 + l;
    const float* bn = Wneigh + col0 + l;

    v8f accS = {};
    v8f accN = {};

#pragma unroll
    for (int kk = 0; kk < IN_FEATS / 4; ++kk) {
        const int k = kk * 4 + half * 2;
        v2f a;
        a.x = arow[kk * 4 + 0];
        a.y = arow[kk * 4 + 1];
        v2f bS;
        bS.x = bs[(size_t)(k + 0) * HID];
        bS.y = bs[(size_t)(k + 1) * HID];
        v2f bN;
        bN.x = bn[(size_t)(k + 0) * HID];
        bN.y = bn[(size_t)(k + 1) * HID];
        // (neg_a, A, neg_b, B, c_mod, C, reuse_a, reuse_b)
        accS = __builtin_amdgcn_wmma_f32_16x16x4_f32(false, a, false, bS,
                                                     (short)0, accS, false, false);
        accN = __builtin_amdgcn_wmma_f32_16x16x4_f32(false, a, false, bN,
                                                     (short)0, accN, false, false);
    }

    // C/D layout: VGPR j holds row M = j + 8*half, col N = lane&15
#pragma unroll
    for (int j = 0; j < 8; ++j) {
        const int row = row0 + j + half * 8;
        const int col = col0 + l;
        xWself[(size_t)row * HID + col] = accS[j];
        hneigh[(size_t)row * HID + col] = accN[j];
    }
}

// -----------------------------------------------------------------------------
// Kernel 2: zero agg + deg (graph-capture safe, no hipMemset)
// -----------------------------------------------------------------------------
__global__ void sage_zero(float* __restrict__ agg, float* __restrict__ deg,
                          int n_agg, int n_deg) {
    const int i = blockIdx.x * blockDim.x + threadIdx.x;
    if (i < n_agg) agg[i] = 0.0f;
    if (i < n_deg) deg[i] = 0.0f;
}

// -----------------------------------------------------------------------------
// Kernel 3: scatter-add messages + degree. 64 threads per edge, one feature
// each. agg (25.6 MB) lives in L2, so these are on-chip f32 atomics.
// -----------------------------------------------------------------------------
__global__ void sage_scatter(const float* __restrict__ hneigh,
                             const int* __restrict__ src,
                             const int* __restrict__ dst,
                             float* __restrict__ agg,
                             float* __restrict__ deg,
                             int n_edges) {
    const long long idx = (long long)blockIdx.x * blockDim.x + threadIdx.x;
    const int e = (int)(idx >> 6);
    const int f = (int)(idx & 63);
    if (e >= n_edges) return;
    const int s = src[e];
    const int d = dst[e];
    const float v = hneigh[(size_t)s * HID + f];
    unsafeAtomicAdd(&agg[(size_t)d * HID + f], v);
    if (f == 0) unsafeAtomicAdd(&deg[d], 1.0f);
}

// -----------------------------------------------------------------------------
// Kernel 4: h = relu(xWself + agg/max(deg,1) + b)   (h overwrites hneigh buf)
// -----------------------------------------------------------------------------
__global__ void sage_finalize(const float* __restrict__ xWself,
                              const float* __restrict__ agg,
                              const float* __restrict__ deg,
                              const float* __restrict__ bias,
                              float* __restrict__ h,
                              long long total) {
    const long long idx = (long long)blockIdx.x * blockDim.x + threadIdx.x;
    if (idx >= total) return;
    const int i = (int)(idx >> 6);
    const int f = (int)(idx & 63);
    const float m = agg[idx] / fmaxf(deg[i], 1.0f);
    const float v = xWself[idx] + m + bias[f];
    h[idx] = fmaxf(v, 0.0f);
}

// -----------------------------------------------------------------------------
// Kernel 5: per-edge dot products. One wave32 per edge, float2 per lane
// (coalesced 256B row reads; h is L2-resident), 5-step shfl_xor reduction.
// Computes pos and neg score for the same edge index.
// -----------------------------------------------------------------------------
__global__ void sage_edge_dot(const float* __restrict__ h,
                              const int* __restrict__ src,
                              const int* __restrict__ dst,
                              const int* __restrict__ nsrc,
                              const int* __restrict__ ndst,
                              float* __restrict__ pos,
                              float* __restrict__ neg,
                              int n_edges) {
    const int lane = threadIdx.x & 31;
    const int e = blockIdx.x * (blockDim.x >> 5) + (threadIdx.x >> 5);
    if (e >= n_edges) return;

    {
        const int u = src[e];
        const int v = dst[e];
        const float2* hu = (const float2*)(h + (size_t)u * HID);
        const float2* hv = (const float2*)(h + (size_t)v * HID);
        const float2 a = hu[lane];
        const float2 b = hv[lane];
        float p = a.x * b.x + a.y * b.y;
#pragma unroll
        for (int off = 16; off >= 1; off >>= 1) p += __shfl_xor(p, off, 32);
        if (lane == 0) pos[e] = p;
    }
    {
        const int u = nsrc[e];
        const int v = ndst[e];
        const float2* hu = (const float2*)(h + (size_t)u * HID);
        const float2* hv = (const float2*)(h + (size_t)v * HID);
        const float2 a = hu[lane];
        const float2 b = hv[lane];
        float p = a.x * b.x + a.y * b.y;
#pragma unroll
        for (int off = 16; off >= 1; off >>= 1) p += __shfl_xor(p, off, 32);
        if (lane == 0) neg[e] = p;
    }
}

// -----------------------------------------------------------------------------
// Launch
// -----------------------------------------------------------------------------
extern "C" void kernel_launch(void* const* d_in, const int* in_sizes, int n_in,
                              void* d_out, int out_size, void* d_ws, size_t ws_size,
                              hipStream_t stream) {
    const float* x      = (const float*)d_in[0];
    const float* Wself  = (const float*)d_in[1];
    const float* Wneigh = (const float*)d_in[2];
    const float* bias   = (const float*)d_in[3];
    const int*   src    = (const int*)d_in[4];
    const int*   dst    = (const int*)d_in[5];
    const int*   nsrc   = (const int*)d_in[6];
    const int*   ndst   = (const int*)d_in[7];

    const int n_nodes = in_sizes[0] / IN_FEATS;   // 100000
    const int n_edges = in_sizes[4];              // 1600000

    float* out = (float*)d_out;
    float* pos = out;
    float* neg = out + n_edges;

    // Workspace layout (fp32): hneigh | xWself | agg | deg  (~77 MB)
    float* ws = (float*)d_ws;
    const size_t nh = (size_t)n_nodes * HID;
    float* hneigh = ws;
    float* xWself = ws + nh;
    float* agg    = ws + 2 * nh;
    float* deg    = ws + 3 * nh;

    // 1) dual GEMM (16 rows / block, 4 waves of one 16-col tile each)
    sage_dual_gemm<<<n_nodes / 16, 128, 0, stream>>>(x, Wself, Wneigh, xWself, hneigh);

    // 2) zero accumulators
    {
        const int zn = (int)nh;
        sage_zero<<<(zn + 255) / 256, 256, 0, stream>>>(agg, deg, zn, n_nodes);
    }

    // 3) scatter-add (64 threads per edge)
    {
        const long long t = (long long)n_edges * HID;
        sage_scatter<<<(int)((t + 255) / 256), 256, 0, stream>>>(hneigh, src, dst,
                                                                 agg, deg, n_edges);
    }

    // 4) mean + self term + bias + relu  (h overwrites hneigh buffer)
    {
        const long long t = (long long)nh;
        sage_finalize<<<(int)((t + 255) / 256), 256, 0, stream>>>(xWself, agg, deg,
                                                                  bias, hneigh, t);
    }

    // 5) edge dots: one wave per edge, 8 edges per 256-thread block
    sage_edge_dot<<<(n_edges + 7) / 8, 256, 0, stream>>>(hneigh, src, dst, nsrc, ndst,
                                                         pos, neg, n_edges);
}